// _PointnetSAModuleBase_4209067950518
// MI455X (gfx1250) — compile-verified
//
#include <hip/hip_runtime.h>

#define Bn 8
#define Nn 16384
#define Cn 16
#define Pn 1024
#define Sn 64
#define R2c 0.04f
#define PPB 8

typedef float v2f __attribute__((ext_vector_type(2)));
typedef float v8f __attribute__((ext_vector_type(8)));

__device__ __forceinline__ v8f wmma4(v2f a, v2f b, v8f c) {
  // V_WMMA_F32_16X16X4_F32 : D = A(16x4) * B(4x16) + C(16x16), fp32
  return __builtin_amdgcn_wmma_f32_16x16x4_f32(false, a, false, b, (short)0, c,
                                               false, false);
}

// ---------------------------------------------------------------------------
// Kernel 1: furthest point sampling. 1 block per batch, 1024 threads,
// each thread owns 16 points (strided, coalesced). Writes new_xyz directly.
// ---------------------------------------------------------------------------
__global__ __launch_bounds__(1024) void fps_kernel(
    const float* __restrict__ xyz, float* __restrict__ new_xyz) {
  const int b = blockIdx.x;
  const int t = threadIdx.x;
  const float* xb = xyz + (size_t)b * Nn * 3;

  float px[16], py[16], pz[16], dmin[16];
#pragma unroll
  for (int k = 0; k < 16; ++k) {
    int i = t + k * 1024;
    px[k] = xb[i * 3 + 0];
    py[k] = xb[i * 3 + 1];
    pz[k] = xb[i * 3 + 2];
    dmin[k] = 1e10f;
  }

  __shared__ float s_val[1024];
  __shared__ int s_idx[1024];
  __shared__ float s_pt[3];

  if (t == 0) {
    s_pt[0] = xb[0];
    s_pt[1] = xb[1];
    s_pt[2] = xb[2];
    new_xyz[((size_t)b * Pn + 0) * 3 + 0] = xb[0];
    new_xyz[((size_t)b * Pn + 0) * 3 + 1] = xb[1];
    new_xyz[((size_t)b * Pn + 0) * 3 + 2] = xb[2];
  }
  __syncthreads();

  for (int j = 1; j < Pn; ++j) {
    const float qx = s_pt[0], qy = s_pt[1], qz = s_pt[2];
    float bestv = -1.0f;
    int besti = 0;
#pragma unroll
    for (int k = 0; k < 16; ++k) {
      float dx = px[k] - qx, dy = py[k] - qy, dz = pz[k] - qz;
      float d = dx * dx + dy * dy + dz * dz;
      float dm = fminf(dmin[k], d);
      dmin[k] = dm;
      if (dm > bestv) {  // strict: keeps lowest index within thread
        bestv = dm;
        besti = t + k * 1024;
      }
    }
    s_val[t] = bestv;
    s_idx[t] = besti;
    __syncthreads();
    for (int s = 512; s > 0; s >>= 1) {
      if (t < s) {
        float ov = s_val[t + s];
        int oi = s_idx[t + s];
        float mv = s_val[t];
        int mi = s_idx[t];
        if (ov > mv || (ov == mv && oi < mi)) {
          s_val[t] = ov;
          s_idx[t] = oi;
        }
      }
      __syncthreads();
    }
    if (t == 0) {
      int li = s_idx[0];
      float x0 = xb[li * 3 + 0], y0 = xb[li * 3 + 1], z0 = xb[li * 3 + 2];
      s_pt[0] = x0;
      s_pt[1] = y0;
      s_pt[2] = z0;
      new_xyz[((size_t)b * Pn + j) * 3 + 0] = x0;
      new_xyz[((size_t)b * Pn + j) * 3 + 1] = y0;
      new_xyz[((size_t)b * Pn + j) * 3 + 2] = z0;
    }
    __syncthreads();
  }
}

// ---------------------------------------------------------------------------
// Kernel 2: ball query. One wave32 per query point; ballot + prefix popcount
// appends in-ball indices in ascending order (first-64 semantics).
// ---------------------------------------------------------------------------
__global__ __launch_bounds__(256) void ballq_kernel(
    const float* __restrict__ xyz, const float* __restrict__ new_xyz,
    int* __restrict__ idxout) {
  const int gw = (int)((blockIdx.x * blockDim.x + threadIdx.x) >> 5);
  const int lane = threadIdx.x & 31;
  const int b = gw / Pn;
  const float* xb = xyz + (size_t)b * Nn * 3;
  const float cx = new_xyz[gw * 3 + 0];
  const float cy = new_xyz[gw * 3 + 1];
  const float cz = new_xyz[gw * 3 + 2];
  int* out = idxout + (size_t)gw * Sn;

  int cnt = 0;
  for (int base = 0; base < Nn && cnt < Sn; base += 32) {
    const int i = base + lane;
    float dx = xb[i * 3 + 0] - cx;
    float dy = xb[i * 3 + 1] - cy;
    float dz = xb[i * 3 + 2] - cz;
    float d2 = dx * dx + dy * dy + dz * dz;
    bool inb = d2 < R2c;
    unsigned long long mb = __ballot(inb);
    if (inb) {
      int pos = cnt + (int)__popcll(mb & ((1ull << lane) - 1ull));
      if (pos < Sn) out[pos] = i;
    }
    cnt += (int)__popcll(mb);
  }
  if (cnt > Sn) cnt = Sn;
  if (cnt < Sn) {  // query point itself guarantees cnt >= 1
    int first = 0;
    if (lane == 0) first = out[0];
    first = __shfl(first, 0, 32);
    for (int i2 = cnt + lane; i2 < Sn; i2 += 32) out[i2] = first;
  }
}

// ---------------------------------------------------------------------------
// Kernel 3: gather + 3-layer MLP (19->64->64->128) + max over 64 samples.
// 128 threads (4 waves) per block, wave w owns sample rows w*16..w*16+15.
// fp32 WMMA 16x16x4; activations staged in LDS with conflict-free strides.
// ---------------------------------------------------------------------------
__global__ __launch_bounds__(128) void mlp_kernel(
    const float* __restrict__ xyz, const float* __restrict__ feat,
    const float* __restrict__ W1, const float* __restrict__ b1,
    const float* __restrict__ W2, const float* __restrict__ b2,
    const float* __restrict__ W3, const float* __restrict__ b3,
    const float* __restrict__ new_xyz, const int* __restrict__ idxbuf,
    float* __restrict__ outf) {
  __shared__ __attribute__((aligned(16))) float sW1[64 * 20];   // K padded 19->20
  __shared__ __attribute__((aligned(16))) float sW2[64 * 68];   // stride 68
  __shared__ __attribute__((aligned(16))) float sX[4][16 * 34]; // stride 34
  __shared__ __attribute__((aligned(16))) float sH[4][16 * 68]; // stride 68
  __shared__ __attribute__((aligned(16))) float smax[4][128];

  const int tid = threadIdx.x;
  const int w = tid >> 5;
  const int lane = tid & 31;
  const int ml = lane & 15;     // row (A) or col (B/D) within tile
  const int hi = lane >> 4;     // half select
  const int kh = hi * 2;        // K sub-offset for 16x16x4 fragments

  // --- preload W1 (zero-padded) and W2 into LDS ---
  for (int i = tid; i < 64 * 20; i += 128) {
    int r = i / 20, c = i - r * 20;
    sW1[i] = (c < 19) ? W1[r * 19 + c] : 0.f;
  }
  for (int i = tid; i < 64 * 64; i += 128) {
    int r = i >> 6, c = i & 63;
    sW2[r * 68 + c] = W2[i];
  }
  __syncthreads();

  for (int it = 0; it < PPB; ++it) {
    const int gp = blockIdx.x * PPB + it;
    const int b = gp / Pn;
    const int p = gp - b * Pn;

    // --- gather this wave's 16-sample x 19-channel tile into LDS ---
    const float cx = new_xyz[gp * 3 + 0];
    const float cy = new_xyz[gp * 3 + 1];
    const float cz = new_xyz[gp * 3 + 2];
    {
      const int m = ml;
      const int j = idxbuf[(size_t)gp * Sn + w * 16 + m];
      const float* fp = feat + ((size_t)b * Nn + j) * Cn;
      float* xr = &sX[w][m * 34];
      if (hi == 0) {
        const float* pp = xyz + ((size_t)b * Nn + j) * 3;
        xr[0] = pp[0] - cx;
        xr[1] = pp[1] - cy;
        xr[2] = pp[2] - cz;
        xr[19] = 0.f;  // K padding
        for (int c = 0; c < 8; ++c) xr[3 + c] = fp[c];
      } else {
        for (int c = 8; c < 16; ++c) xr[3 + c] = fp[c];
      }
    }
    __syncthreads();  // also guards smax reuse from previous iteration

    // --- layer 1: X(16x20) @ W1^T -> relu -> sH (16x64) ---
    for (int nt = 0; nt < 4; ++nt) {
      v8f acc = {0.f, 0.f, 0.f, 0.f, 0.f, 0.f, 0.f, 0.f};
      const float bv = b1[nt * 16 + ml];
      const float* arow = &sX[w][ml * 34 + kh];
      const float* bcol = &sW1[(nt * 16 + ml) * 20 + kh];
      for (int kt = 0; kt < 5; ++kt)
        acc = wmma4(*(const v2f*)(arow + kt * 4), *(const v2f*)(bcol + kt * 4), acc);
      float* hr = &sH[w][nt * 16 + ml];
#pragma unroll
      for (int r = 0; r < 8; ++r)
        hr[(r + 8 * hi) * 68] = fmaxf(acc[r] + bv, 0.f);
    }

    // --- layer 2: h1(16x64) @ W2^T -> relu -> sH (buffer all tiles first) ---
    v8f acc2[4];
    for (int nt = 0; nt < 4; ++nt) {
      v8f acc = {0.f, 0.f, 0.f, 0.f, 0.f, 0.f, 0.f, 0.f};
      const float* arow = &sH[w][ml * 68 + kh];
      const float* bcol = &sW2[(nt * 16 + ml) * 68 + kh];
      for (int kt = 0; kt < 16; ++kt)
        acc = wmma4(*(const v2f*)(arow + kt * 4), *(const v2f*)(bcol + kt * 4), acc);
      acc2[nt] = acc;
    }
    for (int nt = 0; nt < 4; ++nt) {
      const float bv = b2[nt * 16 + ml];
      float* hr = &sH[w][nt * 16 + ml];
#pragma unroll
      for (int r = 0; r < 8; ++r)
        hr[(r + 8 * hi) * 68] = fmaxf(acc2[nt][r] + bv, 0.f);
    }

    // --- layer 3: h2(16x64) @ W3^T -> relu -> max over 16 rows per wave ---
    for (int nt = 0; nt < 8; ++nt) {
      v8f acc = {0.f, 0.f, 0.f, 0.f, 0.f, 0.f, 0.f, 0.f};
      const float bv = b3[nt * 16 + ml];
      const float* arow = &sH[w][ml * 68 + kh];
      const float* bcol = W3 + (size_t)(nt * 16 + ml) * 64 + kh;  // L2-resident
      for (int kt = 0; kt < 16; ++kt)
        acc = wmma4(*(const v2f*)(arow + kt * 4), *(const v2f*)(bcol + kt * 4), acc);
      float cm = -1e30f;
#pragma unroll
      for (int r = 0; r < 8; ++r)
        cm = fmaxf(cm, fmaxf(acc[r] + bv, 0.f));
      cm = fmaxf(cm, __shfl_xor(cm, 16, 32));  // merge rows m and m+8 halves
      if (hi == 0) smax[w][nt * 16 + ml] = cm;
    }
    __syncthreads();

    // --- reduce across the 4 waves, write (B,128,P) layout ---
    const float m0 = fmaxf(fmaxf(smax[0][tid], smax[1][tid]),
                           fmaxf(smax[2][tid], smax[3][tid]));
    outf[(size_t)b * 128 * Pn + (size_t)tid * Pn + p] = m0;
  }
}

// ---------------------------------------------------------------------------
extern "C" void kernel_launch(void* const* d_in, const int* in_sizes, int n_in,
                              void* d_out, int out_size, void* d_ws,
                              size_t ws_size, hipStream_t stream) {
  const float* xyz = (const float*)d_in[0];
  const float* feat = (const float*)d_in[1];
  const float* W1 = (const float*)d_in[2];
  const float* b1 = (const float*)d_in[3];
  const float* W2 = (const float*)d_in[4];
  const float* b2 = (const float*)d_in[5];
  const float* W3 = (const float*)d_in[6];
  const float* b3 = (const float*)d_in[7];

  float* newxyz = (float*)d_out;                      // (B, P, 3)
  float* outf = (float*)d_out + (size_t)Bn * Pn * 3;  // (B, 128, P)
  int* idxb = (int*)d_ws;                             // (B, P, 64)

  fps_kernel<<<Bn, 1024, 0, stream>>>(xyz, newxyz);
  ballq_kernel<<<(Bn * Pn * 32) / 256, 256, 0, stream>>>(xyz, newxyz, idxb);
  mlp_kernel<<<(Bn * Pn) / PPB, 128, 0, stream>>>(xyz, feat, W1, b1, W2, b2,
                                                  W3, b3, newxyz, idxb, outf);
}